// HeteroGIN_49976239456902
// MI455X (gfx1250) — compile-verified
//
#include <hip/hip_runtime.h>

// ---------------------------------------------------------------------------
// HeteroGIN forward for MI455X (gfx1250, wave32).
//   init:  xu = x_user @ Wi0 + bi0 ; xi = x_item @ Wi1 + bi1
//   per layer: agg = segment_sum(gather) ; fused (1+eps)x+agg -> MLP(2x Lin+ReLU)
//              -> LayerNorm -> ReLU, all in one kernel (weights in LDS).
// Matrix math: V_WMMA_F32_16X16X4_F32 (exact fp32 on the CDNA5 matrix pipe).
// ---------------------------------------------------------------------------

#define NU     100000
#define NI     50000
#define CDIM   128
#define NEDGE  600000
#define WS     132   // padded LDS row stride for weights (bank-conflict free)
#define AS     132   // padded LDS row stride for activations

typedef __attribute__((ext_vector_type(2))) float v2f;
typedef __attribute__((ext_vector_type(8))) float v8f;

// ---- wave-level (32x128)@(128x16) fp32 WMMA micro-kernel -------------------
// Fragment layout assumptions (ISA 7.12.2, 32-bit operands, wave32):
//   A 16x4 : lane<16 holds {A[m][k0],A[m][k0+1]}, lane>=16 holds {A[m][k0+2],A[m][k0+3]}
//   B 4x16 : lane<16 holds {B[k0][n],B[k0+1][n]}, lane>=16 holds {B[k0+2][n],B[k0+3][n]}
//   C/D    : vgpr r = D[r + 8*(lane>>4)][n], n = lane&15 (per 16x16 tile)
__device__ __forceinline__ void wave_gemm_32x16(const float* __restrict__ Alds,
                                                const float* __restrict__ Wlds,
                                                int ncol, int lane,
                                                v8f& acc0, v8f& acc1)
{
  const int hk  = lane >> 4;
  const int l16 = lane & 15;
  const float* ar0 = Alds + l16 * AS;          // rows 0..15 of the 32-row tile
  const float* ar1 = Alds + (l16 + 16) * AS;   // rows 16..31
  const float* bc  = Wlds + ncol + l16;        // this wave's 16-col strip
  v8f z = {};
  acc0 = z; acc1 = z;
#pragma unroll
  for (int k0 = 0; k0 < CDIM; k0 += 4) {
    const int ka = k0 + 2 * hk;
    v2f a0 = { ar0[ka], ar0[ka + 1] };
    v2f a1 = { ar1[ka], ar1[ka + 1] };
    v2f b  = { bc[ka * WS], bc[(ka + 1) * WS] };
    acc0 = __builtin_amdgcn_wmma_f32_16x16x4_f32(false, a0, false, b, (short)0,
                                                 acc0, false, false);
    acc1 = __builtin_amdgcn_wmma_f32_16x16x4_f32(false, a1, false, b, (short)0,
                                                 acc1, false, false);
  }
}

__device__ __forceinline__ void load_weights_lds(float* Wlds, const float* __restrict__ Wg,
                                                 int tid, const float* __restrict__ pf)
{
  // 128x128 f32 -> LDS with padded stride; optionally prefetch next weight mat.
  for (int i = tid; i < (CDIM * CDIM) / 4; i += 256) {
    float4 w = ((const float4*)Wg)[i];
    int r = i >> 5;            // (i*4)/128
    int c = (i & 31) * 4;      // (i*4)%128
    *(float4*)&Wlds[r * WS + c] = w;
    if (pf) __builtin_prefetch(pf + i * 4, 0, 0);   // global_prefetch_b8
  }
}

// ---- init projection: Y = X @ W + b ---------------------------------------
__global__ __launch_bounds__(256) void gemm_init_kernel(
    const float* __restrict__ X, const float* __restrict__ Wg,
    const float* __restrict__ bg, float* __restrict__ Y, int nrows)
{
  __shared__ float Wlds[CDIM * WS];
  __shared__ float Albuf[32 * AS];

  const int tid  = threadIdx.x;
  const int lane = tid & 31, wave = tid >> 5;
  const int rowbase = blockIdx.x * 32;

  load_weights_lds(Wlds, Wg, tid, nullptr);

  const int arow = tid >> 3;
  const int acol = (tid & 7) * 16;
  const int grow = rowbase + arow;
  const float* xr = X + (size_t)grow * CDIM + acol;
#pragma unroll
  for (int j = 0; j < 4; ++j) {
    float4 t4 = make_float4(0.f, 0.f, 0.f, 0.f);
    if (grow < nrows) t4 = *(const float4*)(xr + 4 * j);
    *(float4*)&Albuf[arow * AS + acol + 4 * j] = t4;
  }
  __syncthreads();

  v8f acc0, acc1;
  wave_gemm_32x16(Albuf, Wlds, wave * 16, lane, acc0, acc1);

  const int l16 = lane & 15, hk = lane >> 4;
  const int col = wave * 16 + l16;
  const float bias = bg[col];
#pragma unroll
  for (int r = 0; r < 8; ++r) {
    int r0 = rowbase + r + 8 * hk;
    int r1 = r0 + 16;
    if (r0 < nrows) Y[(size_t)r0 * CDIM + col] = acc0[r] + bias;
    if (r1 < nrows) Y[(size_t)r1 * CDIM + col] = acc1[r] + bias;
  }
}

// ---- fused GIN node update ------------------------------------------------
// out = relu(LN(relu(relu(((1+eps)x+agg)@W0+b0)@W1+b1) * g + beta))
__global__ __launch_bounds__(256) void gin_update_kernel(
    const float* __restrict__ X, const float* __restrict__ AGG,
    const float* __restrict__ epsp,
    const float* __restrict__ W0, const float* __restrict__ b0,
    const float* __restrict__ W1, const float* __restrict__ b1,
    const float* __restrict__ lng, const float* __restrict__ lnb,
    float* __restrict__ OUT, int nrows)
{
  __shared__ float Wlds[CDIM * WS];   // 67.6 KB (reused for W0 then W1)
  __shared__ float Albuf[32 * AS];    // 16.9 KB (z tile, later reused for new_n)
  __shared__ float Hbuf[32 * AS];     // 16.9 KB (MLP hidden)

  const int tid  = threadIdx.x;
  const int lane = tid & 31, wave = tid >> 5;
  const int rowbase = blockIdx.x * 32;
  const float ep = 1.0f + *epsp;

  // stage 1: W0 -> LDS (prefetch W1), z=(1+eps)x+agg -> LDS
  load_weights_lds(Wlds, W0, tid, W1);
  {
    const int arow = tid >> 3;
    const int acol = (tid & 7) * 16;
    const int grow = rowbase + arow;
    const float* xr = X   + (size_t)grow * CDIM + acol;
    const float* ar = AGG + (size_t)grow * CDIM + acol;
#pragma unroll
    for (int j = 0; j < 4; ++j) {
      float4 t4 = make_float4(0.f, 0.f, 0.f, 0.f);
      if (grow < nrows) {
        float4 xv = *(const float4*)(xr + 4 * j);
        float4 av = *(const float4*)(ar + 4 * j);
        t4 = make_float4(ep * xv.x + av.x, ep * xv.y + av.y,
                         ep * xv.z + av.z, ep * xv.w + av.w);
      }
      *(float4*)&Albuf[arow * AS + acol + 4 * j] = t4;
    }
  }
  __syncthreads();

  const int l16 = lane & 15, hk = lane >> 4;
  const int col = wave * 16 + l16;

  // stage 2: h = relu(z @ W0 + b0) -> Hbuf
  {
    v8f acc0, acc1;
    wave_gemm_32x16(Albuf, Wlds, wave * 16, lane, acc0, acc1);
    const float bb = b0[col];
#pragma unroll
    for (int r = 0; r < 8; ++r) {
      Hbuf[(r + 8 * hk) * AS + col]      = fmaxf(acc0[r] + bb, 0.f);
      Hbuf[(16 + r + 8 * hk) * AS + col] = fmaxf(acc1[r] + bb, 0.f);
    }
  }
  __syncthreads();

  // stage 3: W1 -> LDS
  load_weights_lds(Wlds, W1, tid, nullptr);
  __syncthreads();

  // stage 4: n = relu(h @ W1 + b1) -> Albuf (reused)
  {
    v8f acc0, acc1;
    wave_gemm_32x16(Hbuf, Wlds, wave * 16, lane, acc0, acc1);
    const float bb = b1[col];
#pragma unroll
    for (int r = 0; r < 8; ++r) {
      Albuf[(r + 8 * hk) * AS + col]      = fmaxf(acc0[r] + bb, 0.f);
      Albuf[(16 + r + 8 * hk) * AS + col] = fmaxf(acc1[r] + bb, 0.f);
    }
  }
  __syncthreads();

  // stage 5: LayerNorm over channels + affine + ReLU, one wave per row group
  const float4 g4  = ((const float4*)lng)[lane];
  const float4 be4 = ((const float4*)lnb)[lane];
#pragma unroll
  for (int rr = 0; rr < 4; ++rr) {
    const int row  = rr * 8 + wave;
    const int grow = rowbase + row;
    float4 vv = *(const float4*)&Albuf[row * AS + lane * 4];
    float s  = vv.x + vv.y + vv.z + vv.w;
    float ss = vv.x * vv.x + vv.y * vv.y + vv.z * vv.z + vv.w * vv.w;
#pragma unroll
    for (int m = 16; m >= 1; m >>= 1) {
      s  += __shfl_xor(s, m, 32);
      ss += __shfl_xor(ss, m, 32);
    }
    const float mean = s * (1.0f / CDIM);
    const float var  = ss * (1.0f / CDIM) - mean * mean;
    const float rstd = rsqrtf(var + 1e-5f);
    float4 o;
    o.x = fmaxf((vv.x - mean) * rstd * g4.x + be4.x, 0.f);
    o.y = fmaxf((vv.y - mean) * rstd * g4.y + be4.y, 0.f);
    o.z = fmaxf((vv.z - mean) * rstd * g4.z + be4.z, 0.f);
    o.w = fmaxf((vv.w - mean) * rstd * g4.w + be4.w, 0.f);
    if (grow < nrows)
      *(float4*)&OUT[(size_t)grow * CDIM + lane * 4] = o;
  }
}

// ---- edge gather + atomic scatter-add -------------------------------------
__global__ __launch_bounds__(256) void scatter_add_kernel(
    const float* __restrict__ x, const int* __restrict__ src,
    const int* __restrict__ dst, float* __restrict__ agg, int ne)
{
  const int i = blockIdx.x * blockDim.x + threadIdx.x;   // 64 threads / edge
  const int e = i >> 6;
  if (e >= ne) return;
  const int c = (i & 63) * 2;
  const int s = src[e];
  const int d = dst[e];
  const float2 v = *(const float2*)(x + (size_t)s * CDIM + c);
  atomicAdd(agg + (size_t)d * CDIM + c,     v.x);   // global_atomic_add_f32
  atomicAdd(agg + (size_t)d * CDIM + c + 1, v.y);
}

__global__ __launch_bounds__(256) void zero_kernel(float* __restrict__ p, int n)
{
  const int j = (blockIdx.x * blockDim.x + threadIdx.x) * 4;
  if (j < n) *(float4*)(p + j) = make_float4(0.f, 0.f, 0.f, 0.f);
}

// ---------------------------------------------------------------------------
extern "C" void kernel_launch(void* const* d_in, const int* in_sizes, int n_in,
                              void* d_out, int out_size, void* d_ws, size_t ws_size,
                              hipStream_t stream)
{
  (void)in_sizes; (void)n_in; (void)out_size; (void)ws_size;

  const float* x_user = (const float*)d_in[0];
  const float* x_item = (const float*)d_in[1];
  const float* W_init = (const float*)d_in[2];   // (2,C,C)
  const float* b_init = (const float*)d_in[3];   // (2,C)
  const float* mlp_W  = (const float*)d_in[4];   // (L,2,NN,C,C)
  const float* mlp_b  = (const float*)d_in[5];   // (L,2,NN,C)
  const float* eps    = (const float*)d_in[6];   // (L,2)
  const float* ln_g   = (const float*)d_in[7];   // (L,2,C)
  const float* ln_b   = (const float*)d_in[8];   // (L,2,C)
  const int*   e_ui   = (const int*)d_in[9];     // (2,E): [src_user | dst_item]
  const int*   e_iu   = (const int*)d_in[10];    // (2,E): [src_item | dst_user]

  // workspace: xu | xi | agg_u | agg_i  (~154 MB total)
  float* xu    = (float*)d_ws;
  float* xi    = xu    + (size_t)NU * CDIM;
  float* agg_u = xi    + (size_t)NI * CDIM;
  float* agg_i = agg_u + (size_t)NU * CDIM;

  float* out_u = (float*)d_out;
  float* out_i = out_u + (size_t)NU * CDIM;

  const dim3 blk(256);
  const int gu = (NU + 31) / 32;     // 3125
  const int gi = (NI + 31) / 32;     // 1563
  const int gsc = NEDGE / 4;         // 64 threads/edge, 256/block
  const int gzu = (NU * CDIM / 4) / 256;
  const int gzi = (NI * CDIM / 4) / 256;

  // initial per-type projection
  gemm_init_kernel<<<gu, blk, 0, stream>>>(x_user, W_init,                b_init,        xu, NU);
  gemm_init_kernel<<<gi, blk, 0, stream>>>(x_item, W_init + CDIM * CDIM,  b_init + CDIM, xi, NI);

  for (int l = 0; l < 2; ++l) {
    zero_kernel<<<gzi, blk, 0, stream>>>(agg_i, NI * CDIM);
    zero_kernel<<<gzu, blk, 0, stream>>>(agg_u, NU * CDIM);
    scatter_add_kernel<<<gsc, blk, 0, stream>>>(xu, e_ui, e_ui + NEDGE, agg_i, NEDGE);
    scatter_add_kernel<<<gsc, blk, 0, stream>>>(xi, e_iu, e_iu + NEDGE, agg_u, NEDGE);

    // mlp_W[l][t][i] : (((l*2 + t)*2 + i) * C * C)
    const float* Wi0 = mlp_W + ((size_t)(l * 2 + 0) * 2 + 0) * CDIM * CDIM;  // item path lin0
    const float* Wi1 = mlp_W + ((size_t)(l * 2 + 0) * 2 + 1) * CDIM * CDIM;  // item path lin1
    const float* Wu0 = mlp_W + ((size_t)(l * 2 + 1) * 2 + 0) * CDIM * CDIM;  // user path lin0
    const float* Wu1 = mlp_W + ((size_t)(l * 2 + 1) * 2 + 1) * CDIM * CDIM;  // user path lin1
    const float* bi0 = mlp_b + ((size_t)(l * 2 + 0) * 2 + 0) * CDIM;
    const float* bi1 = mlp_b + ((size_t)(l * 2 + 0) * 2 + 1) * CDIM;
    const float* bu0 = mlp_b + ((size_t)(l * 2 + 1) * 2 + 0) * CDIM;
    const float* bu1 = mlp_b + ((size_t)(l * 2 + 1) * 2 + 1) * CDIM;
    const float* eps_i = eps + l * 2 + 0;   // edge type 0: user->item
    const float* eps_u = eps + l * 2 + 1;   // edge type 1: item->user
    const float* g_u = ln_g + (size_t)(l * 2 + 0) * CDIM;  // node type 0 = user
    const float* b_u = ln_b + (size_t)(l * 2 + 0) * CDIM;
    const float* g_i = ln_g + (size_t)(l * 2 + 1) * CDIM;  // node type 1 = item
    const float* b_i = ln_b + (size_t)(l * 2 + 1) * CDIM;

    const bool last = (l == 1);
    float* dst_i = last ? out_i : xi;   // in-place safe: blocks touch only own rows
    float* dst_u = last ? out_u : xu;

    gin_update_kernel<<<gi, blk, 0, stream>>>(xi, agg_i, eps_i, Wi0, bi0, Wi1, bi1,
                                              g_i, b_i, dst_i, NI);
    gin_update_kernel<<<gu, blk, 0, stream>>>(xu, agg_u, eps_u, Wu0, bu0, Wu1, bu1,
                                              g_u, b_u, dst_u, NU);
  }
}